// ConditionalInstanceNorm1D_21260088115456
// MI455X (gfx1250) — compile-verified
//
#include <hip/hip_runtime.h>
#include <stdint.h>

// Conditional InstanceNorm1D  (x:[B=32, C=256, L=8192] fp32)
// HBM-bound kernel: read x once (async global->LDS, NT), write out once (NT).
// One workgroup owns ROWS_PER_BLOCK full rows; double-buffered async staging.

typedef float v4f __attribute__((ext_vector_type(4)));

#define L_DIM          8192
#define C_DIM          256
#define THREADS        256
#define ROWS_PER_BLOCK 8
#define V4_PER_THREAD  (L_DIM / (THREADS * 4))   // 8 float4 per thread per row
#define NWAVES         (THREADS / 32)            // 8 waves (wave32)

// CDNA5 async global->LDS copy, 16B per lane, non-temporal (bypass-ish) load.
// VGLOBAL async encoding: vdst = LDS byte address, vaddr = 64-bit global addr.
__device__ __forceinline__ void async_copy_b128(const float* g, uint32_t lds_byte)
{
    asm volatile("global_load_async_to_lds_b128 %0, %1, off th:TH_LOAD_NT"
                 :: "v"(lds_byte), "v"((unsigned long long)(uintptr_t)g)
                 : "memory");
}

__global__ __launch_bounds__(THREADS)
void cinorm1d_kernel(const float* __restrict__ x,
                     const float* __restrict__ weight,
                     const float* __restrict__ bias,
                     const int*   __restrict__ style_ids,
                     float*       __restrict__ out)
{
    __shared__ __align__(16) float buf[2][L_DIM];     // 2 x 32KB row buffers
    __shared__ float s_psum[NWAVES], s_psq[NWAVES];
    __shared__ float s_scale, s_shift;

    const int  t    = threadIdx.x;
    const int  lane = t & 31;
    const int  wid  = t >> 5;
    const long row0 = (long)blockIdx.x * ROWS_PER_BLOCK;

    // Prologue: stage row 0 into buf[0] (8 async b128 per thread -> ASYNCcnt=8)
    {
        const float*   src = x + row0 * L_DIM;
        const uint32_t dst = (uint32_t)(uintptr_t)&buf[0][0];
#pragma unroll
        for (int j = 0; j < V4_PER_THREAD; ++j) {
            const int e = (j * THREADS + t) * 4;     // float index, 16B aligned
            async_copy_b128(src + e, dst + (uint32_t)e * 4u);
        }
    }

    for (int r = 0; r < ROWS_PER_BLOCK; ++r) {
        const long row = row0 + r;

        // Issue next row's async loads, then wait for the *current* row's 8
        // (async loads complete in order -> ASYNCcnt<=8 means first 8 done).
        if (r + 1 < ROWS_PER_BLOCK) {
            const float*   src = x + (row + 1) * L_DIM;
            const uint32_t dst = (uint32_t)(uintptr_t)&buf[(r + 1) & 1][0];
#pragma unroll
            for (int j = 0; j < V4_PER_THREAD; ++j) {
                const int e = (j * THREADS + t) * 4;
                async_copy_b128(src + e, dst + (uint32_t)e * 4u);
            }
            asm volatile("s_wait_asynccnt 0x8" ::: "memory");
        } else {
            asm volatile("s_wait_asynccnt 0x0" ::: "memory");
        }
        __syncthreads();   // all waves waited -> buf[r&1] fully resident

        // Pass 1: LDS -> registers, accumulate sum / sumsq.
        const v4f* lbuf = (const v4f*)&buf[r & 1][0];
        v4f  v[V4_PER_THREAD];
        float sum = 0.0f, sq = 0.0f;
#pragma unroll
        for (int j = 0; j < V4_PER_THREAD; ++j) {
            v4f q = lbuf[j * THREADS + t];
            v[j] = q;
            sum += (q.x + q.y) + (q.z + q.w);
            sq   = fmaf(q.x, q.x, fmaf(q.y, q.y, fmaf(q.z, q.z, fmaf(q.w, q.w, sq))));
        }

        // wave32 butterfly reduction, then cross-wave combine in LDS.
#pragma unroll
        for (int off = 16; off > 0; off >>= 1) {
            sum += __shfl_xor(sum, off, 32);
            sq  += __shfl_xor(sq,  off, 32);
        }
        if (lane == 0) { s_psum[wid] = sum; s_psq[wid] = sq; }
        __syncthreads();   // (also orders all buf reads before next-iter issues)

        if (t == 0) {
            float S = 0.0f, Q = 0.0f;
#pragma unroll
            for (int wv = 0; wv < NWAVES; ++wv) { S += s_psum[wv]; Q += s_psq[wv]; }
            const float invL = 1.0f / (float)L_DIM;
            float mean = S * invL;
            float var  = fmaxf(fmaf(-mean, mean, Q * invL), 0.0f);  // biased var
            float ve   = var + 1e-5f;
            float inv  = rsqrtf(ve);
            inv = inv * fmaf(-0.5f * ve * inv, inv, 1.5f);          // Newton step
            const int b   = (int)(row >> 8);            // row / C_DIM
            const int c   = (int)(row & (C_DIM - 1));   // row % C_DIM
            const int sid = style_ids[b];
            const float w  = weight[sid * C_DIM + c];
            const float bi = bias[sid * C_DIM + c];
            const float a  = inv * w;
            s_scale = a;
            s_shift = fmaf(-mean, a, bi);               // b - mean*inv*w
        }
        __syncthreads();

        // Pass 2: registers -> global, single FMA per element, NT stores.
        const float a = s_scale, sh = s_shift;
        float* dst = out + row * L_DIM;
#pragma unroll
        for (int j = 0; j < V4_PER_THREAD; ++j) {
            const int e = (j * THREADS + t) * 4;
            v4f q = v[j], o;
            o.x = fmaf(q.x, a, sh);
            o.y = fmaf(q.y, a, sh);
            o.z = fmaf(q.z, a, sh);
            o.w = fmaf(q.w, a, sh);
            __builtin_nontemporal_store(o, (v4f*)(dst + e));
        }
        // No trailing barrier needed: all buf[r&1] reads happened before the
        // post-reduction __syncthreads above; next overwrite of this buffer is
        // issued only after every thread passes it.
    }
}

extern "C" void kernel_launch(void* const* d_in, const int* in_sizes, int n_in,
                              void* d_out, int out_size, void* d_ws, size_t ws_size,
                              hipStream_t stream)
{
    const float* x   = (const float*)d_in[0];   // [B, C, L] fp32
    const float* w   = (const float*)d_in[1];   // [S, C]    fp32
    const float* b   = (const float*)d_in[2];   // [S, C]    fp32
    const int*   sid = (const int*)  d_in[3];   // [B]       int32
    float*       out = (float*)d_out;           // [B, C, L] fp32

    const int rows = in_sizes[0] / L_DIM;       // B*C = 8192
    const int grid = rows / ROWS_PER_BLOCK;     // 1024 blocks of 256 threads

    cinorm1d_kernel<<<dim3(grid), dim3(THREADS), 0, stream>>>(x, w, b, sid, out);
}